// CNAPSProtoNetSimilarityModule_88622355185857
// MI455X (gfx1250) — compile-verified
//
#include <hip/hip_runtime.h>
#include <hip/hip_bf16.h>

typedef __attribute__((ext_vector_type(2))) float v2f;
typedef __attribute__((ext_vector_type(8))) float v8f;

constexpr int kD = 512;   // feature dim
constexpr int kB = 128;   // episodes
constexpr int kL = 512;   // pad length

// ---------------------------------------------------------------- zero output
__global__ void zero_out_kernel(float* out, int n) {
    int i = blockIdx.x * blockDim.x + threadIdx.x;
    if (i < n) out[i] = 0.0f;
}

// ------------------------------------------------- batch start via bin search
__global__ void bstart_kernel(const int* __restrict__ bidx, int N, int* bstart) {
    int b = blockIdx.x * blockDim.x + threadIdx.x;
    if (b > kB) return;
    if (b == kB) { bstart[b] = N; return; }
    int lo = 0, hi = N;
    while (lo < hi) { int mid = (lo + hi) >> 1; if (bidx[mid] < b) lo = mid + 1; else hi = mid; }
    bstart[b] = lo;
}

// ------------------------------------- stable per-batch support/query ranking
// sup_info packs ((flat_row * kD) << 1) | label  -> single-load row descriptor
__global__ void __launch_bounds__(256)
classify_kernel(const int* __restrict__ is_query, const int* __restrict__ labels,
                const int* __restrict__ bstart,
                int* __restrict__ sup_info, int* __restrict__ qry_rows,
                int* __restrict__ meta) {
    int b = blockIdx.x, t = threadIdx.x;
    for (int i = t; i < kL; i += 256) { sup_info[b * kL + i] = 0; qry_rows[b * kL + i] = -1; }
    __shared__ int scan[256];
    __shared__ int sc_carry, q_carry, nposc, nnegc;
    if (t == 0) { sc_carry = 0; q_carry = 0; nposc = 0; nnegc = 0; }
    __syncthreads();
    int start = bstart[b], end = bstart[b + 1];
    for (int base = start; base < end; base += 256) {
        int n = base + t;
        bool valid = n < end;
        int q = valid ? is_query[n] : 0;
        scan[t] = q;
        __syncthreads();
        for (int off = 1; off < 256; off <<= 1) {
            int add = (t >= off) ? scan[t - off] : 0;
            __syncthreads();
            scan[t] += add;
            __syncthreads();
        }
        int qbefore = scan[t] - q;          // stable #queries before me in chunk
        int totq = scan[255];
        int sc = sc_carry, qc = q_carry;
        if (valid) {
            if (q) {
                int rk = qc + qbefore;
                if (rk < kL) qry_rows[b * kL + rk] = n;
            } else {
                int rk = sc + (t - qbefore);
                if (rk < kL) {
                    int lab = (labels[n] == 1) ? 1 : 0;
                    sup_info[b * kL + rk] = ((n * kD) << 1) | lab;
                    if (lab) atomicAdd(&nposc, 1); else atomicAdd(&nnegc, 1);
                }
            }
        }
        __syncthreads();
        if (t == 0) {
            int cn = min(256, end - base);
            q_carry += totq;
            sc_carry += cn - totq;
        }
        __syncthreads();
    }
    if (t == 0) {
        meta[b * 8 + 0] = min(sc_carry, kL);   // stored support rows
        meta[b * 8 + 1] = sc_carry;            // full support length
        meta[b * 8 + 2] = min(q_carry, kL);    // stored query rows
        meta[b * 8 + 3] = q_carry;             // full query length
        meta[b * 8 + 4] = nposc;               // stored positive rows
        meta[b * 8 + 5] = nnegc;               // stored negative rows
    }
}

// ------------------------------------------------------------ per-class means
__global__ void __launch_bounds__(512)
means_kernel(const float* __restrict__ gr, const int* __restrict__ sup_info,
             const int* __restrict__ meta, float* __restrict__ means) {
    int b = blockIdx.x, d = threadIdx.x;
    int scnt = meta[b * 8 + 0];
    float slen = (float)meta[b * 8 + 1];
    float npos = (float)meta[b * 8 + 4];
    float nneg = (float)meta[b * 8 + 5];
    const int* info = sup_info + b * kL;
    float sa = 0.f, sp = 0.f, sn = 0.f;
    for (int i = 0; i < scnt; ++i) {
        int j = info[i];
        float x = gr[(j >> 1) + d];
        sa += x;
        if (j & 1) sp += x; else sn += x;
    }
    means[(size_t)(0 * kB + b) * kD + d] = sa / slen;
    means[(size_t)(1 * kB + b) * kD + d] = sp / npos;
    means[(size_t)(2 * kB + b) * kD + d] = sn / nneg;
}

// ---------------- covariance GEMM via V_WMMA_F32_16X16X4_F32
// One wave computes a 16x32 tile with 6 accumulators (task/pos/neg x 2 cols).
// Branch-free main loop (all rows valid); single masked tail step.
// Emits the regularized systems M = 0.1*label_cov + 0.9*task_cov + 0.1*I.
__global__ void __launch_bounds__(256)
cov_gemm_kernel(const float* __restrict__ gr, const int* __restrict__ sup_info,
                const int* __restrict__ meta, const float* __restrict__ means,
                float* __restrict__ Msys) {
    int b = blockIdx.y;
    int wave = threadIdx.x >> 5;
    int lane = threadIdx.x & 31;
    int tile = blockIdx.x * 8 + wave;            // 0..511 over (D/16)x(D/32)
    int d0 = (tile >> 4) << 4;                   // row block (16)
    int e0 = (tile & 15) << 5;                   // col block (32)

    int scnt = meta[b * 8 + 0];
    float slen = (float)meta[b * 8 + 1];
    float npos = (float)meta[b * 8 + 4];
    float nneg = (float)meta[b * 8 + 5];

    const float* tmean = means + (size_t)(0 * kB + b) * kD;
    const float* pmean = means + (size_t)(1 * kB + b) * kD;
    const float* nmean = means + (size_t)(2 * kB + b) * kD;

    int m  = lane & 15;
    int kh = (lane >> 4) << 1;                   // 0 or 2 (K half per lane group)
    int dA  = d0 + m;
    int dB0 = e0 + m;
    int dB1 = e0 + 16 + m;

    float tmA  = tmean[dA],  pmA  = pmean[dA],  nmA  = nmean[dA];
    float tmB0 = tmean[dB0], pmB0 = pmean[dB0], nmB0 = nmean[dB0];
    float tmB1 = tmean[dB1], pmB1 = pmean[dB1], nmB1 = nmean[dB1];

    v8f ct0 = {}, cp0 = {}, cn0 = {};
    v8f ct1 = {}, cp1 = {}, cn1 = {};
    const int* info = sup_info + b * kL;

    int kfull = scnt & ~3;                       // fully-valid 4-row steps
    int i0 = 0, i1 = 0;
    if (kfull > 0) { i0 = info[kh]; i1 = info[kh + 1]; }
    for (int k = 0; k < kfull; k += 4) {
        int pf = (k + 4 + kh) & (kL - 1);        // even -> pf+1 in range
        int ni0 = info[pf], ni1 = info[pf + 1];  // prefetch next row infos
        int f0 = i0 >> 1, f1 = i1 >> 1;          // element offsets (row*kD)
        int l0 = i0 & 1,  l1 = i1 & 1;
        float xa0 = gr[f0 + dA],  xa1 = gr[f1 + dA];
        float xb0 = gr[f0 + dB0], xb1 = gr[f1 + dB0];
        float xc0 = gr[f0 + dB1], xc1 = gr[f1 + dB1];

        v2f at = { xa0 - tmA, xa1 - tmA };
        v2f ap = { l0 ? xa0 - pmA : 0.f, l1 ? xa1 - pmA : 0.f };
        v2f an = { l0 ? 0.f : xa0 - nmA, l1 ? 0.f : xa1 - nmA };
        v2f bt = { xb0 - tmB0, xb1 - tmB0 };
        v2f bp = { l0 ? xb0 - pmB0 : 0.f, l1 ? xb1 - pmB0 : 0.f };
        v2f bn = { l0 ? 0.f : xb0 - nmB0, l1 ? 0.f : xb1 - nmB0 };
        v2f qt = { xc0 - tmB1, xc1 - tmB1 };
        v2f qp = { l0 ? xc0 - pmB1 : 0.f, l1 ? xc1 - pmB1 : 0.f };
        v2f qn = { l0 ? 0.f : xc0 - nmB1, l1 ? 0.f : xc1 - nmB1 };

        ct0 = __builtin_amdgcn_wmma_f32_16x16x4_f32(false, at, false, bt, (short)0, ct0, false, false);
        cp0 = __builtin_amdgcn_wmma_f32_16x16x4_f32(false, ap, false, bp, (short)0, cp0, false, false);
        cn0 = __builtin_amdgcn_wmma_f32_16x16x4_f32(false, an, false, bn, (short)0, cn0, false, false);
        ct1 = __builtin_amdgcn_wmma_f32_16x16x4_f32(false, at, false, qt, (short)0, ct1, false, false);
        cp1 = __builtin_amdgcn_wmma_f32_16x16x4_f32(false, ap, false, qp, (short)0, cp1, false, false);
        cn1 = __builtin_amdgcn_wmma_f32_16x16x4_f32(false, an, false, qn, (short)0, cn1, false, false);
        i0 = ni0; i1 = ni1;
    }

    if (kfull < scnt) {                          // masked tail, no divergence
        int r0 = kfull + kh, r1 = r0 + 1;
        bool v0 = r0 < scnt, v1 = r1 < scnt;
        int j0 = info[r0 & (kL - 1)], j1 = info[r1 & (kL - 1)];
        int f0 = j0 >> 1, f1 = j1 >> 1;
        int l0 = j0 & 1,  l1 = j1 & 1;
        float xa0 = gr[f0 + dA],  xa1 = gr[f1 + dA];
        float xb0 = gr[f0 + dB0], xb1 = gr[f1 + dB0];
        float xc0 = gr[f0 + dB1], xc1 = gr[f1 + dB1];

        v2f at = { v0 ? xa0 - tmA : 0.f, v1 ? xa1 - tmA : 0.f };
        v2f ap = { (v0 && l0) ? xa0 - pmA : 0.f, (v1 && l1) ? xa1 - pmA : 0.f };
        v2f an = { (v0 && !l0) ? xa0 - nmA : 0.f, (v1 && !l1) ? xa1 - nmA : 0.f };
        v2f bt = { v0 ? xb0 - tmB0 : 0.f, v1 ? xb1 - tmB0 : 0.f };
        v2f bp = { (v0 && l0) ? xb0 - pmB0 : 0.f, (v1 && l1) ? xb1 - pmB0 : 0.f };
        v2f bn = { (v0 && !l0) ? xb0 - nmB0 : 0.f, (v1 && !l1) ? xb1 - nmB0 : 0.f };
        v2f qt = { v0 ? xc0 - tmB1 : 0.f, v1 ? xc1 - tmB1 : 0.f };
        v2f qp = { (v0 && l0) ? xc0 - pmB1 : 0.f, (v1 && l1) ? xc1 - pmB1 : 0.f };
        v2f qn = { (v0 && !l0) ? xc0 - nmB1 : 0.f, (v1 && !l1) ? xc1 - nmB1 : 0.f };

        ct0 = __builtin_amdgcn_wmma_f32_16x16x4_f32(false, at, false, bt, (short)0, ct0, false, false);
        cp0 = __builtin_amdgcn_wmma_f32_16x16x4_f32(false, ap, false, bp, (short)0, cp0, false, false);
        cn0 = __builtin_amdgcn_wmma_f32_16x16x4_f32(false, an, false, bn, (short)0, cn0, false, false);
        ct1 = __builtin_amdgcn_wmma_f32_16x16x4_f32(false, at, false, qt, (short)0, ct1, false, false);
        cp1 = __builtin_amdgcn_wmma_f32_16x16x4_f32(false, ap, false, qp, (short)0, cp1, false, false);
        cn1 = __builtin_amdgcn_wmma_f32_16x16x4_f32(false, an, false, qn, (short)0, cn1, false, false);
    }

    float it  = 1.f / (slen - 1.f);
    float ip  = 1.f / (npos - 1.f);
    float inn = 1.f / (nneg - 1.f);

    int eh = lane >> 4;
    int ec0 = e0 + (lane & 15);
    int ec1 = e0 + 16 + (lane & 15);
    float* Mp = Msys + (size_t)(b * 2 + 0) * kD * kD;
    float* Mn = Msys + (size_t)(b * 2 + 1) * kD * kD;
    for (int r = 0; r < 8; ++r) {
        int drow = d0 + r + eh * 8;
        float tv0 = 0.9f * ct0[r] * it;
        float tv1 = 0.9f * ct1[r] * it;
        float dg0 = (drow == ec0) ? 0.1f : 0.f;
        float dg1 = (drow == ec1) ? 0.1f : 0.f;
        Mp[(size_t)drow * kD + ec0] = 0.1f * cp0[r] * ip  + tv0 + dg0;
        Mn[(size_t)drow * kD + ec0] = 0.1f * cn0[r] * inn + tv0 + dg0;
        Mp[(size_t)drow * kD + ec1] = 0.1f * cp1[r] * ip  + tv1 + dg1;
        Mn[(size_t)drow * kD + ec1] = 0.1f * cn1[r] * inn + tv1 + dg1;
    }
}

// ---------------------------------- CG solve: M w = 1 (w = rowsum of inv(M))
__global__ void __launch_bounds__(256)
cg_solve_kernel(const float* __restrict__ Msys, float* __restrict__ wvec) {
    int sys = blockIdx.x;                         // 0..2B-1
    const float* M = Msys + (size_t)sys * kD * kD;
    __shared__ float xv[kD], rv[kD], pv[kD], Av[kD];
    __shared__ float red[256];
    __shared__ float sh_rs, sh_pap, sh_beta;
    int t = threadIdx.x, lane = t & 31, wave = t >> 5;
    for (int i = t; i < kD; i += 256) { xv[i] = 0.f; rv[i] = 1.f; pv[i] = 1.f; }
    if (t == 0) sh_rs = (float)kD;
    __syncthreads();
    const float4* pv4 = (const float4*)pv;
    for (int it = 0; it < 48; ++it) {
        // Av = M * pv : wave handles 64 contiguous rows, float4 per lane
        for (int j = 0; j < 64; ++j) {
            int row = wave * 64 + j;
            const float4* Mr4 = (const float4*)(M + (size_t)row * kD);
            float s = 0.f;
            for (int c = lane; c < kD / 4; c += 32) {
                float4 mv = Mr4[c];
                float4 p4 = pv4[c];
                s += mv.x * p4.x + mv.y * p4.y + mv.z * p4.z + mv.w * p4.w;
            }
            for (int msk = 16; msk > 0; msk >>= 1) s += __shfl_xor(s, msk, 32);
            if (lane == 0) Av[row] = s;
        }
        __syncthreads();
        float s = pv[t] * Av[t] + pv[t + 256] * Av[t + 256];
        red[t] = s; __syncthreads();
        for (int off = 128; off > 0; off >>= 1) { if (t < off) red[t] += red[t + off]; __syncthreads(); }
        if (t == 0) sh_pap = red[0];
        __syncthreads();
        float alpha = sh_rs / sh_pap;
        for (int i = t; i < kD; i += 256) { xv[i] += alpha * pv[i]; rv[i] -= alpha * Av[i]; }
        __syncthreads();
        float s2 = rv[t] * rv[t] + rv[t + 256] * rv[t + 256];
        red[t] = s2; __syncthreads();
        for (int off = 128; off > 0; off >>= 1) { if (t < off) red[t] += red[t + off]; __syncthreads(); }
        if (t == 0) { sh_beta = red[0] / sh_rs; sh_rs = red[0]; }
        __syncthreads();
        float beta = sh_beta;
        for (int i = t; i < kD; i += 256) pv[i] = rv[i] + beta * pv[i];
        __syncthreads();
    }
    for (int i = t; i < kD; i += 256) wvec[(size_t)sys * kD + i] = xv[i];
}

// ------------------------------------------- logits: (diff.w) * (diff.1) * s
__global__ void __launch_bounds__(128)
logits_kernel(const float* __restrict__ gr, const float* __restrict__ log_scale,
              const int* __restrict__ qry_rows, const int* __restrict__ meta,
              const float* __restrict__ means, const float* __restrict__ wvec,
              float* __restrict__ out) {
    int gid = blockIdx.x;                       // b*kL + i
    int b = gid >> 9;
    int i = gid & (kL - 1);
    int qlen = meta[b * 8 + 3];
    if (i >= qlen) return;                      // masked rows stay zeroed
    int qcnt = meta[b * 8 + 2];
    int flat = (i < qcnt) ? qry_rows[b * kL + i] : -1;
    const float* pmean = means + (size_t)(1 * kB + b) * kD;
    const float* nmean = means + (size_t)(2 * kB + b) * kD;
    const float* wp = wvec + (size_t)(b * 2 + 0) * kD;
    const float* wn = wvec + (size_t)(b * 2 + 1) * kD;
    int t = threadIdx.x;
    float ap = 0.f, sp = 0.f, an = 0.f, sn = 0.f;
    for (int d = t; d < kD; d += 128) {
        float x = (flat >= 0) ? gr[(size_t)flat * kD + d] : 0.f;
        float dp = x - pmean[d], dn = x - nmean[d];
        ap += dp * wp[d]; sp += dp;
        an += dn * wn[d]; sn += dn;
    }
    __shared__ float r0[128], r1[128], r2[128], r3[128];
    r0[t] = ap; r1[t] = sp; r2[t] = an; r3[t] = sn;
    __syncthreads();
    for (int off = 64; off > 0; off >>= 1) {
        if (t < off) { r0[t] += r0[t + off]; r1[t] += r1[t + off]; r2[t] += r2[t + off]; r3[t] += r3[t + off]; }
        __syncthreads();
    }
    if (t == 0) {
        float sc = __expf(log_scale[0]);
        out[(size_t)gid * 2 + 0] = r2[0] * r3[0] * sc;   // neg_maha
        out[(size_t)gid * 2 + 1] = r0[0] * r1[0] * sc;   // pos_maha
    }
}

// ---------------------------------------------------------------------- glue
extern "C" void kernel_launch(void* const* d_in, const int* in_sizes, int n_in,
                              void* d_out, int out_size, void* d_ws, size_t ws_size,
                              hipStream_t stream) {
    const float* gr        = (const float*)d_in[0];
    const float* log_scale = (const float*)d_in[1];
    const int*   labels    = (const int*)d_in[2];
    const int*   is_query  = (const int*)d_in[3];
    const int*   bidx      = (const int*)d_in[4];
    float*       out       = (float*)d_out;
    const int N = in_sizes[2];

    char* base = (char*)d_ws;
    size_t off = 0;
    auto alloc = [&](size_t bytes) { size_t o = off; off = (off + bytes + 255) & ~(size_t)255; return o; };
    int*   bstart   = (int*)(base + alloc((size_t)(kB + 1) * 4));
    int*   sup_info = (int*)(base + alloc((size_t)kB * kL * 4));
    int*   qry_rows = (int*)(base + alloc((size_t)kB * kL * 4));
    int*   meta     = (int*)(base + alloc((size_t)kB * 8 * 4));
    float* means    = (float*)(base + alloc((size_t)3 * kB * kD * 4));
    float* wvec     = (float*)(base + alloc((size_t)2 * kB * kD * 4));
    float* Msys     = (float*)(base + alloc((size_t)2 * kB * kD * kD * 4));
    (void)ws_size; (void)n_in;

    zero_out_kernel<<<(out_size + 255) / 256, 256, 0, stream>>>(out, out_size);
    bstart_kernel<<<1, 256, 0, stream>>>(bidx, N, bstart);
    classify_kernel<<<kB, 256, 0, stream>>>(is_query, labels, bstart, sup_info, qry_rows, meta);
    means_kernel<<<kB, kD, 0, stream>>>(gr, sup_info, meta, means);
    cov_gemm_kernel<<<dim3(64, kB), 256, 0, stream>>>(gr, sup_info, meta, means, Msys);
    cg_solve_kernel<<<2 * kB, 256, 0, stream>>>(Msys, wvec);
    logits_kernel<<<kB * kL, 128, 0, stream>>>(gr, log_scale, qry_rows, meta, means, wvec, out);
}